// Classifier_41583873360056
// MI455X (gfx1250) — compile-verified
//
#include <hip/hip_runtime.h>

// ---------------------------------------------------------------------------
// GNN classifier for MI455X (gfx1250, wave32).
//
// Input order (setup_inputs dict order; params flattened jax-style, sorted
// keys: Wout, bout, layers[ W1,W2,b1,b2,bt,bt1,g,g1 ]):
//   d_in[0] seq1  (50000*128 f32)
//   d_in[1] rows  (800000 int)
//   d_in[2] cols  (800000 int)
//   d_in[3] Wout  (128 f32)
//   d_in[4] bout  (1 f32)
//   d_in[5 + 8*l + {0..7}] = W1, W2, b1, b2, bt, bt1, g, g1 for layer l
// ---------------------------------------------------------------------------

#define HID       128
#define N_LAYERS  3
#define BN_EPS    1e-5f

typedef __attribute__((ext_vector_type(2))) float v2f;
typedef __attribute__((ext_vector_type(8))) float v8f;

// ---------------------------------------------------------------------------
// Zero-fill
// ---------------------------------------------------------------------------
__global__ void zero_kernel(float* __restrict__ p, size_t n) {
  size_t i = (size_t)blockIdx.x * blockDim.x + threadIdx.x;
  if (i < n) p[i] = 0.0f;
}

// ---------------------------------------------------------------------------
// Segment-sum: pooled[rows[e]][c] += h[cols[e]][c].
// One lane per (edge, feature): gathers are coalesced 512B rows, scatter uses
// native f32 global atomics (destination is L2-resident in MI455X's 192MB L2).
// e = gid>>7 is wave-uniform (waves are 32-aligned, 128 | gid granularity).
// ---------------------------------------------------------------------------
__global__ __launch_bounds__(256) void scatter_add_kernel(
    float* __restrict__ pooled, const float* __restrict__ h,
    const int* __restrict__ rows, const int* __restrict__ cols, int nEdges) {
  size_t gid = (size_t)blockIdx.x * 256 + threadIdx.x;
  size_t e = gid >> 7;
  int c = (int)(gid & 127);
  if (e >= (size_t)nEdges) return;
  int dst = rows[e];
  int src = cols[e];
  float v = h[(size_t)src * HID + c];
  atomicAdd(pooled + (size_t)dst * HID + c, v);
}

// ---------------------------------------------------------------------------
// Y[Npad x 128] = X[Npad x 128] @ W[128 x 128] + bias, via fp32 WMMA
// (V_WMMA_F32_16X16X4_F32, full fp32 precision as the reference requires).
//
// Block = 256 threads (8 waves); each wave owns TWO 16-row strips (32 rows x
// 128 cols, 16 v8f accumulators) so each set of B-fragment LDS reads feeds
// 16 WMMAs instead of 8 (LDS issue rate, not the matrix pipe, bounds the
// inner loop otherwise).
//
// W is staged in 64KB LDS via CDNA5 async global->LDS B128 copies (ASYNCcnt),
// with a per-row rotate-by-16*(k&7) swizzle: B-frag reads hit rows k (lanes
// 0-15) and k+2 (lanes 16-31); raw stride-128 rows alias the same 64 banks,
// the rotation puts the two half-waves on disjoint 16-bank windows
// (16*k vs 16*(k+2) = +32 mod 64). Rotation by multiples of 16 keeps 16-byte
// chunks contiguous, so B128 async-copy granularity is preserved.
// ---------------------------------------------------------------------------
__global__ __launch_bounds__(256) void gemm_bias_wmma_kernel(
    float* __restrict__ Y, const float* __restrict__ X,
    const float* __restrict__ W, const float* __restrict__ bias) {
  __shared__ float wlds[HID * HID];  // 64 KB

  const int tid = threadIdx.x;

  // Async-stage W (swizzled) into LDS: 16 B128 copies per lane, no VGPR
  // round-trip; wait on ASYNCcnt, then barrier.
  {
    const unsigned ldsBase = (unsigned)(size_t)(&wlds[0]);
    for (int i = tid * 4; i < HID * HID; i += 256 * 4) {
      int k = i >> 7;   // row
      int c = i & 127;  // col, multiple of 4
      int cs = (c + 16 * (k & 7)) & 127;
      unsigned lofs = ldsBase + (unsigned)(k * HID + cs) * 4u;
      unsigned long long gaddr =
          (unsigned long long)(const void*)(W + k * HID + c);
      asm volatile("global_load_async_to_lds_b128 %0, %1, off"
                   :: "v"(lofs), "v"(gaddr)
                   : "memory");
    }
    asm volatile("s_wait_asynccnt 0x0" ::: "memory");
  }
  __syncthreads();

  const int lane  = tid & 31;
  const int wave  = tid >> 5;
  const int m0    = blockIdx.x * 256 + wave * 32;  // 32-row strip base
  const int mrow  = lane & 15;                     // A-frag: M = lane&15
  const int khalf = (lane >> 4) * 2;               // A/B-frag: K sub-offset
  const int ncol  = lane & 15;                     // B/D-frag: N = lane&15

  v8f acc0[8], acc1[8];
#pragma unroll
  for (int t = 0; t < 8; ++t) {
    acc0[t] = {};
    acc1[t] = {};
  }

  const float* xrow0 = X + (size_t)(m0 + mrow) * HID;
  const float* xrow1 = xrow0 + 16 * HID;

  for (int k = 0; k < HID; k += 4) {
    const int kb = k + khalf;
    // A fragments (16x4 f32): lane holds X[m][kb], X[m][kb+1]; kb even => b64.
    v2f a0 = *reinterpret_cast<const v2f*>(xrow0 + kb);
    v2f a1 = *reinterpret_cast<const v2f*>(xrow1 + kb);

    // B fragments (4x16 f32), shared by both row strips.
    v2f b[8];
#pragma unroll
    for (int t = 0; t < 8; ++t) {
      const int n = t * 16 + ncol;
      b[t].x = wlds[(kb + 0) * HID + ((n + 16 * ((kb + 0) & 7)) & 127)];
      b[t].y = wlds[(kb + 1) * HID + ((n + 16 * ((kb + 1) & 7)) & 127)];
    }
#pragma unroll
    for (int t = 0; t < 8; ++t)
      acc0[t] = __builtin_amdgcn_wmma_f32_16x16x4_f32(
          false, a0, false, b[t], (short)0, acc0[t], false, false);
#pragma unroll
    for (int t = 0; t < 8; ++t)
      acc1[t] = __builtin_amdgcn_wmma_f32_16x16x4_f32(
          false, a1, false, b[t], (short)0, acc1[t], false, false);
  }

  // D layout: VGPR v holds row (v + 8*(lane>>4)), col = lane&15 (+16*t).
  const int mtop = (lane >> 4) * 8;
#pragma unroll
  for (int t = 0; t < 8; ++t) {
    const int n = t * 16 + ncol;
    const float bv = bias[n];
#pragma unroll
    for (int v = 0; v < 8; ++v) {
      Y[(size_t)(m0 + mtop + v) * HID + n]      = acc0[t][v] + bv;
      Y[(size_t)(m0 + 16 + mtop + v) * HID + n] = acc1[t][v] + bv;
    }
  }
}

// ---------------------------------------------------------------------------
// Column sums & sum-of-squares over rows [0, nRows): stats[0:128] = sum,
// stats[128:256] = sumsq. Coalesced column reads, LDS fold, atomic finish.
// ---------------------------------------------------------------------------
__global__ __launch_bounds__(256) void colstats_kernel(
    float* __restrict__ stats, const float* __restrict__ X, int nRows,
    int rowsPerBlock) {
  __shared__ float red[512];
  const int tid  = threadIdx.x;
  const int c    = tid & 127;
  const int half = tid >> 7;
  float s = 0.0f, q = 0.0f;
  int r0 = blockIdx.x * rowsPerBlock;
  int r1 = min(r0 + rowsPerBlock, nRows);
  for (int r = r0 + half; r < r1; r += 2) {
    float x = X[(size_t)r * HID + c];
    s += x;
    q += x * x;
  }
  red[tid]       = s;
  red[256 + tid] = q;
  __syncthreads();
  if (tid < 128) {
    atomicAdd(stats + c,       red[tid]       + red[tid + 128]);
    atomicAdd(stats + 128 + c, red[256 + tid] + red[256 + tid + 128]);
  }
}

// ---------------------------------------------------------------------------
// y = relu(gamma * (x - mu) * rsqrt(var + eps) + beta), population variance.
// ---------------------------------------------------------------------------
__global__ __launch_bounds__(256) void bn_relu_kernel(
    float* __restrict__ Y, const float* __restrict__ X,
    const float* __restrict__ stats, const float* __restrict__ gamma,
    const float* __restrict__ beta, int nRows) {
  size_t gid = (size_t)blockIdx.x * 256 + threadIdx.x;
  if (gid >= (size_t)nRows * HID) return;
  int c = (int)(gid & 127);
  float invN = 1.0f / (float)nRows;
  float mu  = stats[c] * invN;
  float var = stats[128 + c] * invN - mu * mu;
  float x = X[gid];
  float y = gamma[c] * (x - mu) * rsqrtf(var + BN_EPS) + beta[c];
  Y[gid] = fmaxf(y, 0.0f);
}

// ---------------------------------------------------------------------------
// out[i] = dot(H[i], Wout) + bout. One wave (32 lanes) per row; shuffle tree.
// ---------------------------------------------------------------------------
__global__ __launch_bounds__(256) void out_kernel(
    float* __restrict__ out, const float* __restrict__ H,
    const float* __restrict__ Wout, const float* __restrict__ bout,
    int nRows) {
  const int lane = threadIdx.x & 31;
  const int row  = blockIdx.x * 8 + (threadIdx.x >> 5);
  if (row >= nRows) return;
  const float* h = H + (size_t)row * HID;
  float s = 0.0f;
#pragma unroll
  for (int i = 0; i < 4; ++i) s += h[lane + 32 * i] * Wout[lane + 32 * i];
#pragma unroll
  for (int off = 16; off > 0; off >>= 1) s += __shfl_down(s, off, 32);
  if (lane == 0) out[row] = s + bout[0];
}

// ---------------------------------------------------------------------------
extern "C" void kernel_launch(void* const* d_in, const int* in_sizes, int n_in,
                              void* d_out, int out_size, void* d_ws,
                              size_t ws_size, hipStream_t stream) {
  const float* seq1 = (const float*)d_in[0];
  const int*   rows = (const int*)d_in[1];
  const int*   cols = (const int*)d_in[2];
  const float* Wout = (const float*)d_in[3];
  const float* bout = (const float*)d_in[4];

  const int N     = in_sizes[0] / HID;          // 50000
  const int nE    = in_sizes[1];                // 800000
  const int Npad  = ((N + 255) / 256) * 256;    // 50176 (GEMM tiles 256 rows)
  const size_t bufElems = (size_t)Npad * HID;
  const size_t bufBytes = bufElems * sizeof(float);

  char*  ws    = (char*)d_ws;
  float* P     = (float*)(ws);                  // pooled / h1 (reused)
  float* T     = (float*)(ws + bufBytes);       // GEMM output
  float* H     = (float*)(ws + 2 * bufBytes);   // layer output
  float* stats = (float*)(ws + 3 * bufBytes);   // 256 floats

  const int zeroGrid = (int)((bufElems + 255) / 256);
  const int gemmGrid = Npad / 256;
  const int scatGrid = (int)(((size_t)nE * HID + 255) / 256);
  const int statGrid = (N + 511) / 512;
  const int elemGrid = (int)(((size_t)N * HID + 255) / 256);

  const float* hprev = seq1;
  for (int l = 0; l < N_LAYERS; ++l) {
    const int base = 5 + 8 * l;
    const float* W1  = (const float*)d_in[base + 0];
    const float* W2  = (const float*)d_in[base + 1];
    const float* b1  = (const float*)d_in[base + 2];
    const float* b2  = (const float*)d_in[base + 3];
    const float* bt  = (const float*)d_in[base + 4];
    const float* bt1 = (const float*)d_in[base + 5];
    const float* g   = (const float*)d_in[base + 6];
    const float* g1  = (const float*)d_in[base + 7];

    // pooled = segment_sum(hprev[cols], rows)
    zero_kernel<<<zeroGrid, 256, 0, stream>>>(P, bufElems);
    zero_kernel<<<1, 256, 0, stream>>>(stats, 256);
    scatter_add_kernel<<<scatGrid, 256, 0, stream>>>(P, hprev, rows, cols, nE);

    // h1 = relu(BN(pooled @ W1 + b1, g1, bt1))   (h1 written back into P)
    gemm_bias_wmma_kernel<<<gemmGrid, 256, 0, stream>>>(T, P, W1, b1);
    colstats_kernel<<<statGrid, 256, 0, stream>>>(stats, T, N, 512);
    bn_relu_kernel<<<elemGrid, 256, 0, stream>>>(P, T, stats, g1, bt1, N);

    // h = relu(BN(h1 @ W2 + b2, g, bt))
    zero_kernel<<<1, 256, 0, stream>>>(stats, 256);
    gemm_bias_wmma_kernel<<<gemmGrid, 256, 0, stream>>>(T, P, W2, b2);
    colstats_kernel<<<statGrid, 256, 0, stream>>>(stats, T, N, 512);
    bn_relu_kernel<<<elemGrid, 256, 0, stream>>>(H, T, stats, g, bt, N);

    hprev = H;
  }

  out_kernel<<<(N + 7) / 8, 256, 0, stream>>>((float*)d_out, H, Wout, bout, N);
}